// AIGEncoder_18743237280080
// MI455X (gfx1250) — compile-verified
//
#include <hip/hip_runtime.h>
#include <math.h>

#define NN 100000
#define NE 1600000
#define NG 64
#define DIM 128

typedef __attribute__((ext_vector_type(2))) float v2f;
typedef __attribute__((ext_vector_type(8))) float v8f;

// ---------------- node encoder: h = emb_type[type] + emb_inv[ninv] ----------
__global__ void k_encode(const int* __restrict__ nt, const int* __restrict__ ninv,
                         const float* __restrict__ et, const float* __restrict__ ei,
                         float* __restrict__ h) {
    int idx = blockIdx.x * blockDim.x + threadIdx.x;
    if (idx >= NN * DIM) return;
    int n = idx >> 7, d = idx & 127;
    h[idx] = et[nt[n] * DIM + d] + ei[ninv[n] * DIM + d];
}

// ---------------- degree / normalization ------------------------------------
__global__ void k_fill(float* __restrict__ p, float v, int n) {
    int i = blockIdx.x * blockDim.x + threadIdx.x;
    if (i < n) p[i] = v;
}
__global__ void k_deg_scatter(const int* __restrict__ dst, float* __restrict__ deg) {
    int e = blockIdx.x * blockDim.x + threadIdx.x;
    if (e < NE) atomicAdd(&deg[dst[e]], 1.0f);
}
__global__ void k_rsqrt(float* __restrict__ p, int n) {
    int i = blockIdx.x * blockDim.x + threadIdx.x;
    if (i < n) p[i] = rsqrtf(p[i]);
}

// ---------------- fp32 WMMA GEMM: xw = h @ W  (N x 128 @ 128 x 128) ---------
// block = 256 threads = 8 wave32; block handles 16 rows x 128 cols.
// Each wave: one 16x16 tile, K accumulated via V_WMMA_F32_16X16X4_F32.
__global__ __launch_bounds__(256) void k_gemm_wmma(const float* __restrict__ h,
                                                   const float* __restrict__ W,
                                                   float* __restrict__ xw) {
    __shared__ float As[16 * DIM];   // 8 KB A-tile in LDS
    const int tid = threadIdx.x;
    const size_t row0 = (size_t)blockIdx.x * 16;

    const float* __restrict__ hrow = h + row0 * DIM;
#pragma unroll
    for (int i = 0; i < 8; ++i) {
        int idx = tid + i * 256;
        As[idx] = hrow[idx];
    }
    __syncthreads();

    const int wave = tid >> 5;        // 0..7 -> which 16-col tile
    const int lane = tid & 31;
    const int half = lane >> 4;       // 0 or 1
    const int mn   = lane & 15;       // A row / B col within tile
    const int col0 = wave * 16;

    v8f c = {};
#pragma unroll 4
    for (int k0 = 0; k0 < DIM; k0 += 4) {
        const int ka = k0 + 2 * half;
        v2f a, b;
        // A 16x4 f32 frag: lanes 0-15 K={k0,k0+1}, lanes 16-31 K={k0+2,k0+3}
        a.x = As[mn * DIM + ka];
        a.y = As[mn * DIM + ka + 1];
        // B 4x16 f32 frag (W row-major [K][N]), mirrored lane split
        b.x = W[(size_t)ka * DIM + col0 + mn];
        b.y = W[(size_t)(ka + 1) * DIM + col0 + mn];
        c = __builtin_amdgcn_wmma_f32_16x16x4_f32(false, a, false, b,
                                                  (short)0, c, false, false);
    }

    float* __restrict__ out = xw + row0 * DIM;
#pragma unroll
    for (int r = 0; r < 8; ++r) {
        int row = r + 8 * half;       // C/D layout: vgpr r, lane half selects M block
        out[row * DIM + col0 + mn] = c[r];
    }
}

// ---------------- self-loop term + bias: agg = xw*dinv^2 + b ----------------
__global__ void k_selfinit(const float* __restrict__ xw, const float* __restrict__ dinv,
                           const float* __restrict__ bias, float* __restrict__ agg) {
    int idx = blockIdx.x * blockDim.x + threadIdx.x;
    if (idx >= NN * DIM) return;
    int n = idx >> 7, d = idx & 127;
    float di = dinv[n];
    agg[idx] = xw[idx] * (di * di) + bias[d];
}

// ---------------- edge scatter: agg[dst] += xw[src] * dinv[s]*dinv[d] -------
__global__ void k_edge_scatter(const int* __restrict__ src, const int* __restrict__ dst,
                               const float* __restrict__ dinv, const float* __restrict__ xw,
                               float* __restrict__ agg) {
    long long t = (long long)blockIdx.x * blockDim.x + threadIdx.x;
    int e = (int)(t >> 5);
    int l = (int)(t & 31);
    if (e >= NE) return;
    int s = src[e], d = dst[e];
    float coef = dinv[s] * dinv[d];
    float4 v = *((const float4*)(xw + (size_t)s * DIM) + l);
    float* ag = agg + (size_t)d * DIM + l * 4;
    atomicAdd(ag + 0, v.x * coef);
    atomicAdd(ag + 1, v.y * coef);
    atomicAdd(ag + 2, v.z * coef);
    atomicAdd(ag + 3, v.w * coef);
}

__global__ void k_relu(float* __restrict__ p, int n) {
    int i = blockIdx.x * blockDim.x + threadIdx.x;
    if (i < n) p[i] = fmaxf(p[i], 0.0f);
}

// ---------------- segment max/sum readout -----------------------------------
__device__ __forceinline__ unsigned enc_f(float f) {
    unsigned u = __float_as_uint(f);
    return (u & 0x80000000u) ? ~u : (u | 0x80000000u);   // order-preserving
}
__device__ __forceinline__ float dec_f(unsigned e) {
    return (e & 0x80000000u) ? __uint_as_float(e ^ 0x80000000u) : __uint_as_float(~e);
}

__global__ void k_rd_init(unsigned* __restrict__ gmax, float* __restrict__ gsum) {
    int i = blockIdx.x * blockDim.x + threadIdx.x;
    if (i >= NG * DIM) return;
    gmax[i] = 0u;        // below enc of any finite float
    gsum[i] = 0.0f;
}
__global__ void k_readout(const float* __restrict__ h, const int* __restrict__ batch,
                          unsigned* __restrict__ gmax, float* __restrict__ gsum) {
    long long t = (long long)blockIdx.x * blockDim.x + threadIdx.x;
    int n = (int)(t >> 5);
    int l = (int)(t & 31);
    if (n >= NN) return;
    int g = batch[n];
    float4 v = *((const float4*)(h + (size_t)n * DIM) + l);
    unsigned* mx = gmax + g * DIM + l * 4;
    float*    sm = gsum + g * DIM + l * 4;
    atomicMax(mx + 0, enc_f(v.x));  atomicAdd(sm + 0, v.x);
    atomicMax(mx + 1, enc_f(v.y));  atomicAdd(sm + 1, v.y);
    atomicMax(mx + 2, enc_f(v.z));  atomicAdd(sm + 2, v.z);
    atomicMax(mx + 3, enc_f(v.w));  atomicAdd(sm + 3, v.w);
}
__global__ void k_final(const unsigned* __restrict__ gmax, const float* __restrict__ gsum,
                        float* __restrict__ out) {
    int i = blockIdx.x * blockDim.x + threadIdx.x;
    if (i >= NG * DIM) return;
    int g = i >> 7, d = i & 127;
    out[g * 2 * DIM + d]       = rintf(dec_f(gmax[i]) * 1000.0f) * 0.001f;
    out[g * 2 * DIM + DIM + d] = rintf(gsum[i]        * 1000.0f) * 0.001f;
}

// ---------------------------------------------------------------------------
extern "C" void kernel_launch(void* const* d_in, const int* in_sizes, int n_in,
                              void* d_out, int out_size, void* d_ws, size_t ws_size,
                              hipStream_t stream) {
    const int*   node_type = (const int*)d_in[0];
    const int*   ninv      = (const int*)d_in[1];
    const int*   edge      = (const int*)d_in[2];   // [2, NE]
    const int*   src       = edge;
    const int*   dst       = edge + NE;
    const int*   batch     = (const int*)d_in[3];
    const float* emb_type  = (const float*)d_in[4];
    const float* emb_inv   = (const float*)d_in[5];
    const float* Wm[3] = { (const float*)d_in[6], (const float*)d_in[8],  (const float*)d_in[10] };
    const float* bm[3] = { (const float*)d_in[7], (const float*)d_in[9],  (const float*)d_in[11] };
    float* out = (float*)d_out;

    // workspace layout (ping-pong node-feature buffers, L2-resident)
    float*    buf0 = (float*)d_ws;                       // h / agg   (51.2 MB)
    float*    buf1 = buf0 + (size_t)NN * DIM;            // xw        (51.2 MB)
    float*    dinv = buf1 + (size_t)NN * DIM;            // NN floats
    unsigned* gmax = (unsigned*)(dinv + NN);             // NG*DIM
    float*    gsum = (float*)(gmax + NG * DIM);          // NG*DIM

    const int elemBlocks = (NN * DIM + 255) / 256;

    // 1) encode
    k_encode<<<elemBlocks, 256, 0, stream>>>(node_type, ninv, emb_type, emb_inv, buf0);

    // 2) normalization coefficients (layer-invariant): dinv = rsqrt(indeg + 1)
    k_fill<<<(NN + 255) / 256, 256, 0, stream>>>(dinv, 1.0f, NN);
    k_deg_scatter<<<(NE + 255) / 256, 256, 0, stream>>>(dst, dinv);
    k_rsqrt<<<(NN + 255) / 256, 256, 0, stream>>>(dinv, NN);

    // 3) three GCN layers
    for (int layer = 0; layer < 3; ++layer) {
        k_gemm_wmma<<<NN / 16, 256, 0, stream>>>(buf0, Wm[layer], buf1);
        k_selfinit<<<elemBlocks, 256, 0, stream>>>(buf1, dinv, bm[layer], buf0);
        k_edge_scatter<<<(int)(((long long)NE * 32 + 255) / 256), 256, 0, stream>>>(
            src, dst, dinv, buf1, buf0);
        if (layer < 2)
            k_relu<<<elemBlocks, 256, 0, stream>>>(buf0, NN * DIM);
    }

    // 4) readout: segment max + sum over batch, concat, round to 3 decimals
    k_rd_init<<<(NG * DIM + 255) / 256, 256, 0, stream>>>(gmax, gsum);
    k_readout<<<(int)(((long long)NN * 32 + 255) / 256), 256, 0, stream>>>(buf0, batch, gmax, gsum);
    k_final<<<(NG * DIM + 255) / 256, 256, 0, stream>>>(gmax, gsum, out);
}